// EDN_Model_80444737454112
// MI455X (gfx1250) — compile-verified
//
#include <hip/hip_runtime.h>
#include <math.h>

// ---------------------------------------------------------------------------
// EDN equivariant GNN for gfx1250 (MI455X).
//
// Key optimizations (derived from the reference dataflow):
//  * Output depends only on ca[0][:, :, 0]  =>  only conv2 paths with o==0
//    are live: (i,f,o) in {(0,0,0),(1,1,0),(2,2,0)}.  12/15 paths eliminated.
//  * Every W3J tensor actually used is diag: W3J[l,0,l] = W3J[0,l,l] =
//    delta/sqrt(2l+1) in the real basis (unitary A => identity preserved).
//  * Dominant FLOPs = radial MLP layer 2: hidden(E,12) @ W2(12,1600) per path.
//    Mapped to V_WMMA_F32_16X16X4_F32: one wave per 16-edge tile, K=12 as
//    3 chained K=4 WMMAs, bias folded into the C accumulator.
//  * ~16 MB of intermediates -> L2 resident (192 MB); HBM not a factor.
// ---------------------------------------------------------------------------

typedef float v2f __attribute__((ext_vector_type(2)));
typedef float v8f __attribute__((ext_vector_type(8)));

#define E_EDGES 24576
#define NN 4096
#define MULT 40
#define N_CA 512
#define N_GR 8

__device__ __forceinline__ float sspf(float v) {
  // softplus(v) - ln(2), numerically stable
  float sp = fmaxf(v, 0.f) + log1pf(expf(-fabsf(v)));
  return sp - 0.6931471805599453f;
}

// ---------------- edge geometry: r, Y1, Y2 ----------------
__global__ void edge_geom(const float* __restrict__ ev, float* __restrict__ r_out,
                          float* __restrict__ y1, float* __restrict__ y2) {
  int e = blockIdx.x * blockDim.x + threadIdx.x;
  if (e >= E_EDGES) return;
  float x = ev[3 * e], y = ev[3 * e + 1], z = ev[3 * e + 2];
  float r = sqrtf(x * x + y * y + z * z);
  r_out[e] = r;
  float inv = 1.f / (r + 1e-9f);
  x *= inv; y *= inv; z *= inv;
  const float s3 = 1.7320508075688772f;
  const float s15 = 3.872983346207417f;
  const float s5 = 2.23606797749979f;
  y1[3 * e] = s3 * x; y1[3 * e + 1] = s3 * y; y1[3 * e + 2] = s3 * z;
  y2[5 * e]     = s15 * x * y;
  y2[5 * e + 1] = s15 * y * z;
  y2[5 * e + 2] = 0.5f * s5 * (3.f * z * z - 1.f);
  y2[5 * e + 3] = s15 * x * z;
  y2[5 * e + 4] = 0.5f * s15 * (x * x - y * y);
}

// ---------------- h = x @ lin1^T / sqrt(4) ----------------
__global__ void lin1_kernel(const float* __restrict__ x, const float* __restrict__ W,
                            float* __restrict__ h) {
  int idx = blockIdx.x * blockDim.x + threadIdx.x;
  if (idx >= NN * MULT) return;
  int n = idx / MULT, u = idx % MULT;
  float acc = 0.f;
  for (int v = 0; v < 4; v++) acc += W[u * 4 + v] * x[n * 4 + v];
  h[idx] = acc * 0.5f;
}

// ---------------- fused radial-MLP + tensor-product conv (WMMA) ----------------
// One wave (32 thr) per 16-edge tile.
//   hidden(16x12) = relu(basis @ W1 + b1)                 (VALU, tiny)
//   R(16x1600)    = hidden @ W2 + b2                      (v_wmma_f32_16x16x4_f32)
//   t[e,u,i]     += R[e,u,v] * xe[e,v,i]                  (LDS atomics)
//   scatter: EXPAND (conv1, di=1): out[dst,u,o] += cw * t[e,u] * Y[e,o]
//            else   (conv2, o=0):  out[dst,u]   += cw * sum_i t[e,u,i]*Y[e,i]
template <int DI, int DF, bool EXPAND>
__global__ __launch_bounds__(32) void conv_wmma(
    const float* __restrict__ xin,   // (N, MULT*DI)
    const float* __restrict__ rbuf,  // (E,)
    const float* __restrict__ Yf,    // (E, DF)  (null when DF==1 -> Y==1)
    const float* __restrict__ W1, const float* __restrict__ b1,
    const float* __restrict__ W2, const float* __restrict__ b2,
    const int* __restrict__ src, const int* __restrict__ dst,
    float* __restrict__ out, int NB, float maxr, float cw,
    int outStride, int colBase) {
  __shared__ float sB[16 * 40];         // radial basis
  __shared__ float sh[16 * 12];         // hidden (A matrix)
  __shared__ float sxe[16 * MULT * DI]; // gathered node features
  __shared__ float st[16 * MULT * DI];  // t accumulator
  __shared__ int ssrc[16], sdst[16];

  const int lane = threadIdx.x;
  const int ebase = blockIdx.x * 16;

  if (lane < 16) { ssrc[lane] = src[ebase + lane]; sdst[lane] = dst[ebase + lane]; }

  // radial Gaussian basis
  const float c0 = maxr / (float)(NB - 1);
  const float isig = (float)(NB - 1) / (0.8f * maxr);
  for (int idx = lane; idx < 16 * NB; idx += 32) {
    int e = idx / NB, j = idx % NB;
    float tt = (rbuf[ebase + e] - (float)j * c0) * isig;
    sB[idx] = expf(-0.5f * tt * tt);
  }
  __syncthreads();

  // hidden = relu(B @ W1 + b1)  (W1: (NB,12) row-major)
  for (int idx = lane; idx < 16 * 12; idx += 32) {
    int e = idx / 12, k = idx % 12;
    float acc = b1[k];
    for (int j = 0; j < NB; j++) acc += sB[e * NB + j] * W1[j * 12 + k];
    sh[idx] = acc > 0.f ? acc : 0.f;
  }
  // gather xe, zero t
  for (int idx = lane; idx < 16 * MULT * DI; idx += 32) {
    int e = idx / (MULT * DI);
    sxe[idx] = xin[ssrc[e] * (MULT * DI) + (idx - e * MULT * DI)];
    st[idx] = 0.f;
  }
  __syncthreads();

  // A fragments: 16x4 f32 layout -> lanes 0-15: M=lane, {K=0,K=1};
  //                                 lanes 16-31: M=lane-16, {K=2,K=3}
  const int khalf = lane >> 4;
  const int m = lane & 15;
  v2f afrag[3];
#pragma unroll
  for (int s = 0; s < 3; s++) {
    afrag[s].x = sh[m * 12 + 4 * s + 2 * khalf];
    afrag[s].y = sh[m * 12 + 4 * s + 2 * khalf + 1];
  }

  // 100 column tiles of 16 over the 1600 (u,v) pairs
  for (int ct = 0; ct < 100; ct++) {
    const int col = ct * 16 + m;
    const float bias = b2[col];
    v8f acc = {bias, bias, bias, bias, bias, bias, bias, bias};
#pragma unroll
    for (int s = 0; s < 3; s++) {
      v2f bfrag;  // B 4x16: VGPR0 rows {0,2}, VGPR1 rows {1,3}, N=lane&15
      bfrag.x = W2[(4 * s + 2 * khalf) * 1600 + col];
      bfrag.y = W2[(4 * s + 2 * khalf + 1) * 1600 + col];
      acc = __builtin_amdgcn_wmma_f32_16x16x4_f32(false, afrag[s], false, bfrag,
                                                  (short)0, acc, false, false);
    }
    const int u = col / MULT, v = col % MULT;
#pragma unroll
    for (int rr = 0; rr < 8; rr++) {
      const int erow = rr + 8 * khalf;  // C: VGPR r -> M=r (lanes<16) / r+8
      const float Rv = acc[rr];
#pragma unroll
      for (int dc = 0; dc < DI; dc++)
        atomicAdd(&st[erow * (MULT * DI) + u * DI + dc],
                  Rv * sxe[erow * (MULT * DI) + v * DI + dc]);
    }
  }
  __syncthreads();

  if (EXPAND) {
    for (int idx = lane; idx < 16 * MULT * DF; idx += 32) {
      int e = idx / (MULT * DF);
      int rem = idx - e * MULT * DF;
      int u = rem / DF, o = rem % DF;
      float yv = (DF == 1) ? 1.f : Yf[(ebase + e) * DF + o];
      float val = cw * st[e * MULT + u] * yv;  // DI==1 here
      atomicAdd(&out[sdst[e] * outStride + u * DF + o], val);
    }
  } else {
    for (int idx = lane; idx < 16 * MULT; idx += 32) {
      int e = idx / MULT, u = idx % MULT;
      float acc2 = 0.f;
#pragma unroll
      for (int i = 0; i < DI; i++) {
        float yv = (DI == 1) ? 1.f : Yf[(ebase + e) * DI + i];
        acc2 += st[e * MULT * DI + u * DI + i] * yv;
      }
      atomicAdd(&out[sdst[e] * outStride + colBase + u], cw * acc2);
    }
  }
}

// ---------------- fused normlayer -> lin2 -> nonlin -> lin3 ----------------
template <int D, int L>
__global__ __launch_bounds__(128) void node_mid(const float* __restrict__ xin,
                                                const float* __restrict__ W2l,
                                                const float* __restrict__ W3l,
                                                float* __restrict__ out) {
  __shared__ float sx[MULT * D], sy[MULT * D], red[128], fac[MULT];
  const int n = blockIdx.x, t = threadIdx.x;
  const int SZ = MULT * D;
  float local = 0.f;
  for (int i = t; i < SZ; i += 128) { float v = xin[n * SZ + i]; sx[i] = v; local += v * v; }
  red[t] = local;
  __syncthreads();
  for (int s = 64; s > 0; s >>= 1) { if (t < s) red[t] += red[t + s]; __syncthreads(); }
  const float scale = rsqrtf(red[0] / (float)SZ + 1e-8f);
  const float inv40 = 0.15811388300841897f;  // 1/sqrt(40)
  for (int i = t; i < SZ; i += 128) {
    int u = i / D, d = i % D;
    float acc = 0.f;
    for (int v = 0; v < MULT; v++) acc += W2l[u * MULT + v] * sx[v * D + d];
    sy[i] = acc * scale * inv40;
  }
  __syncthreads();
  if (L == 0) {
    for (int i = t; i < SZ; i += 128) sy[i] = sspf(sy[i]);
  } else {
    if (t < MULT) {
      float s2 = 1e-8f;
      for (int d = 0; d < D; d++) { float v = sy[t * D + d]; s2 += v * v; }
      float nn = sqrtf(s2);
      fac[t] = sspf(nn) / nn;
    }
    __syncthreads();
    for (int i = t; i < SZ; i += 128) sy[i] *= fac[i / D];
  }
  __syncthreads();
  for (int i = t; i < SZ; i += 128) {
    int u = i / D, d = i % D;
    float acc = 0.f;
    for (int v = 0; v < MULT; v++) acc += W3l[u * MULT + v] * sy[v * D + d];
    out[n * SZ + i] = acc * inv40;
  }
}

// ---------------- CA: gather -> normlayer -> lin40 -> ssp ----------------
__global__ __launch_bounds__(128) void ca_kernel(const float* __restrict__ cat0,
                                                 const int* __restrict__ ca_idx,
                                                 const float* __restrict__ W4,
                                                 float* __restrict__ out0) {
  __shared__ float sx[120], red[128];
  const int b = blockIdx.x, t = threadIdx.x;
  const int n = ca_idx[b];
  float local = 0.f;
  for (int i = t; i < 120; i += 128) { float v = cat0[n * 120 + i]; sx[i] = v; local += v * v; }
  red[t] = local;
  __syncthreads();
  for (int s = 64; s > 0; s >>= 1) { if (t < s) red[t] += red[t + s]; __syncthreads(); }
  const float scale = rsqrtf(red[0] / 120.f + 1e-8f);
  const float inv = 0.09128709291752768f;  // 1/sqrt(120)
  if (t < MULT) {
    float acc = 0.f;
    for (int v = 0; v < 120; v++) acc += W4[t * 120 + v] * sx[v];
    out0[b * MULT + t] = sspf(acc * scale * inv);
  }
}

// ---------------- segment mean + dense head ----------------
__global__ __launch_bounds__(256) void head_kernel(
    const float* __restrict__ out0, const int* __restrict__ ca_idx,
    const int* __restrict__ node_batch, const float* __restrict__ d1W,
    const float* __restrict__ d1b, const float* __restrict__ d2W,
    const float* __restrict__ d2b, const float* __restrict__ d3W,
    const float* __restrict__ d3b, float* __restrict__ dout) {
  __shared__ float ssum[N_GR * MULT], cnt[N_GR], pooled[N_GR * MULT];
  __shared__ float h1[N_GR * 250], h2[N_GR * 150];
  const int t = threadIdx.x;
  for (int i = t; i < N_GR * MULT; i += 256) ssum[i] = 0.f;
  if (t < N_GR) cnt[t] = 0.f;
  __syncthreads();
  for (int idx = t; idx < N_CA * MULT; idx += 256) {
    int i = idx / MULT, u = idx % MULT;
    int g = node_batch[ca_idx[i]];
    atomicAdd(&ssum[g * MULT + u], out0[idx]);
  }
  for (int i = t; i < N_CA; i += 256) atomicAdd(&cnt[node_batch[ca_idx[i]]], 1.f);
  __syncthreads();
  for (int i = t; i < N_GR * MULT; i += 256)
    pooled[i] = ssum[i] / fmaxf(cnt[i / MULT], 1.f);
  __syncthreads();
  for (int idx = t; idx < N_GR * 250; idx += 256) {
    int g = idx / 250, j = idx % 250;
    float acc = d1b[j];
    for (int v = 0; v < MULT; v++) acc += d1W[j * MULT + v] * pooled[g * MULT + v];
    h1[idx] = acc > 0.f ? acc : expm1f(acc);
  }
  __syncthreads();
  for (int idx = t; idx < N_GR * 150; idx += 256) {
    int g = idx / 150, j = idx % 150;
    float acc = d2b[j];
    for (int v = 0; v < 250; v++) acc += d2W[j * 250 + v] * h1[g * 250 + v];
    h2[idx] = acc > 0.f ? acc : expm1f(acc);
  }
  __syncthreads();
  if (t < N_GR) {
    float acc = d3b[0];
    for (int v = 0; v < 150; v++) acc += d3W[v] * h2[t * 150 + v];
    dout[t] = acc;
  }
}

// ---------------------------------------------------------------------------
// Input flattening assumption (setup_inputs dict order, params depth-first):
//  0:x 1:edge_vec 2:lin1
//  3+6l..8+6l  : conv1l{W1,b1,W2,b2}, lin2l, lin3l          (l=0..2)
//  21+4p..24+4p: conv2 path p {W1,b1,W2,b2}, p in insertion order:
//    p0=(0,0,0) p4=(1,1,0) p12=(2,2,0) are the live ones.
//  81:lin40 82:lin41 83:lin42 84:d1W 85:d1b 86:d2W 87:d2b 88:d3W 89:d3b
//  90:edge_index(2,E) 91:ca_idx 92:node_batch
// ---------------------------------------------------------------------------
extern "C" void kernel_launch(void* const* d_in, const int* in_sizes, int n_in,
                              void* d_out, int out_size, void* d_ws, size_t ws_size,
                              hipStream_t stream) {
  (void)in_sizes; (void)n_in; (void)out_size; (void)ws_size;
  const float* x = (const float*)d_in[0];
  const float* ev = (const float*)d_in[1];
  const float* lin1 = (const float*)d_in[2];
  const int* ei = (const int*)d_in[90];
  const int* ca_idx = (const int*)d_in[91];
  const int* nbatch = (const int*)d_in[92];
  const int* src = ei;
  const int* dst = ei + E_EDGES;

  float* ws = (float*)d_ws;
  float* r_buf = ws;                        // E
  float* Y1 = r_buf + E_EDGES;              // 3E
  float* Y2 = Y1 + 3 * E_EDGES;             // 5E
  float* h_node = Y2 + 5 * E_EDGES;         // 40N
  float* o1_0 = h_node + MULT * NN;         // 40N
  float* o1_1 = o1_0 + MULT * NN;           // 120N
  float* o1_2 = o1_1 + 3 * MULT * NN;       // 200N
  float* s0 = o1_2 + 5 * MULT * NN;         // 40N
  float* s1 = s0 + MULT * NN;               // 120N
  float* s2 = s1 + 3 * MULT * NN;           // 200N
  float* cat0 = s2 + 5 * MULT * NN;         // 120N
  float* ca0 = cat0 + 3 * MULT * NN;        // 512*40

  // zero the atomic accumulators (conv1 outputs are contiguous, then cat0)
  hipMemsetAsync(o1_0, 0, (size_t)(9 * MULT * NN) * sizeof(float), stream);
  hipMemsetAsync(cat0, 0, (size_t)(3 * MULT * NN) * sizeof(float), stream);

  edge_geom<<<(E_EDGES + 255) / 256, 256, 0, stream>>>(ev, r_buf, Y1, Y2);
  lin1_kernel<<<(NN * MULT + 255) / 256, 256, 0, stream>>>(x, lin1, h_node);

  const float cw0 = 0.15811388300841897f;   // 1/sqrt(1*40)
  const float cw1 = 0.09128709291752768f;   // 1/sqrt(3*40)
  const float cw2 = 0.07071067811865475f;   // 1/sqrt(5*40)
  const int nt = E_EDGES / 16;

  // conv1 (input l=0 scalars, output l = 0,1,2), nb=20, max_radius=10
  {
    const float *W1, *b1, *W2, *b2;
    W1 = (const float*)d_in[3]; b1 = (const float*)d_in[4];
    W2 = (const float*)d_in[5]; b2 = (const float*)d_in[6];
    conv_wmma<1, 1, true><<<nt, 32, 0, stream>>>(h_node, r_buf, nullptr, W1, b1, W2, b2,
                                                 src, dst, o1_0, 20, 10.f, cw0, MULT, 0);
    W1 = (const float*)d_in[9]; b1 = (const float*)d_in[10];
    W2 = (const float*)d_in[11]; b2 = (const float*)d_in[12];
    conv_wmma<1, 3, true><<<nt, 32, 0, stream>>>(h_node, r_buf, Y1, W1, b1, W2, b2,
                                                 src, dst, o1_1, 20, 10.f, cw1, 3 * MULT, 0);
    W1 = (const float*)d_in[15]; b1 = (const float*)d_in[16];
    W2 = (const float*)d_in[17]; b2 = (const float*)d_in[18];
    conv_wmma<1, 5, true><<<nt, 32, 0, stream>>>(h_node, r_buf, Y2, W1, b1, W2, b2,
                                                 src, dst, o1_2, 20, 10.f, cw2, 5 * MULT, 0);
  }

  // norm -> lin2 -> nonlin -> lin3 per l
  node_mid<1, 0><<<NN, 128, 0, stream>>>(o1_0, (const float*)d_in[7],
                                         (const float*)d_in[8], s0);
  node_mid<3, 1><<<NN, 128, 0, stream>>>(o1_1, (const float*)d_in[13],
                                         (const float*)d_in[14], s1);
  node_mid<5, 2><<<NN, 128, 0, stream>>>(o1_2, (const float*)d_in[19],
                                         (const float*)d_in[20], s2);

  // conv2: only o==0 paths are live -> cat0 slots {0,1,2}; nb=40, max_radius=20
  {
    const float* W1 = (const float*)d_in[21];  // path (0,0,0)
    const float* b1 = (const float*)d_in[22];
    const float* W2 = (const float*)d_in[23];
    const float* b2 = (const float*)d_in[24];
    conv_wmma<1, 1, false><<<nt, 32, 0, stream>>>(s0, r_buf, nullptr, W1, b1, W2, b2,
                                                  src, dst, cat0, 40, 20.f, cw0, 120, 0);
    W1 = (const float*)d_in[37]; b1 = (const float*)d_in[38];  // path (1,1,0)
    W2 = (const float*)d_in[39]; b2 = (const float*)d_in[40];
    conv_wmma<3, 3, false><<<nt, 32, 0, stream>>>(s1, r_buf, Y1, W1, b1, W2, b2,
                                                  src, dst, cat0, 40, 20.f, cw1, 120, 40);
    W1 = (const float*)d_in[69]; b1 = (const float*)d_in[70];  // path (2,2,0)
    W2 = (const float*)d_in[71]; b2 = (const float*)d_in[72];
    conv_wmma<5, 5, false><<<nt, 32, 0, stream>>>(s2, r_buf, Y2, W1, b1, W2, b2,
                                                  src, dst, cat0, 40, 20.f, cw2, 120, 80);
  }

  ca_kernel<<<N_CA, 128, 0, stream>>>(cat0, ca_idx, (const float*)d_in[81], ca0);
  head_kernel<<<1, 256, 0, stream>>>(ca0, ca_idx, nbatch,
                                     (const float*)d_in[84], (const float*)d_in[85],
                                     (const float*)d_in[86], (const float*)d_in[87],
                                     (const float*)d_in[88], (const float*)d_in[89],
                                     (float*)d_out);
}